// AllDistance_46282567581897
// MI455X (gfx1250) — compile-verified
//
#include <hip/hip_runtime.h>
#include <hip/hip_bf16.h>
#include <stdint.h>

// AllDistance: 12 per-row distances between two [N, 4096] fp32 matrices.
// Bandwidth-bound streaming reduction. CDNA5 path: all-up-front async
// global->LDS copies (ASYNCcnt, saddr form, chunk selected by the dual-side
// instruction offset), FIFO drain via s_wait_asynccnt, wave32 butterfly.

#define DDIM    4096
#define TPB     256
#define CHUNK_B 4096                    // bytes per chunk per input (1024 floats)
#define NCHUNK  (DDIM / (TPB * 4))      // 4 chunks per row
#define NACC    11

// Async copy, saddr form. Per the ISA, INST_OFFSET is added to BOTH the LDS
// address (VDST vgpr) and the global address (SADDR + VADDR vgpr), so with
// matching 4 KB chunk strides on both sides one immediate selects the chunk.
template <int OFF>
__device__ __forceinline__ void async_copy_b128(uint32_t lds_byte_off,
                                                uint32_t voff,
                                                const void* sbase) {
  asm volatile("global_load_async_to_lds_b128 %0, %1, %2 offset:%3"
               :
               : "v"(lds_byte_off), "v"(voff), "s"(sbase), "i"(OFF)
               : "memory");
}

template <int N>
__device__ __forceinline__ void wait_async() {
  asm volatile("s_wait_asynccnt %0" :: "i"(N) : "memory");
}

__device__ __forceinline__ void accum4(float acc[NACC], float4 u4, float4 v4) {
  float us[4] = {u4.x, u4.y, u4.z, u4.w};
  float vs[4] = {v4.x, v4.y, v4.z, v4.w};
  #pragma unroll
  for (int k = 0; k < 4; ++k) {
    const float u = us[k];
    const float v = vs[k];
    const float d  = u - v;
    const float ad = fabsf(d);
    acc[0] += ad;                                   // cityblock / braycurtis num
    acc[1] += fabsf(u + v);                         // braycurtis den
    const float den = fabsf(u) + fabsf(v);
    acc[2] += (den > 0.0f) ? ad * __builtin_amdgcn_rcpf(den) : 0.0f;  // canberra
    acc[3]  = fmaxf(acc[3], ad);                    // chebyshev
    acc[4] += u;                                    // sum u
    acc[5] += v;                                    // sum v
    acc[6]  = fmaf(u, u, acc[6]);                   // sum u^2
    acc[7]  = fmaf(v, v, acc[7]);                   // sum v^2
    acc[8]  = fmaf(u, v, acc[8]);                   // sum uv
    acc[9]  = fmaf(d, d, acc[9]);                   // sum d^2
    acc[10] += (u != v) ? 1.0f : 0.0f;              // hamming count
  }
}

__global__ __launch_bounds__(TPB)
void AllDistance_46282567581897_kernel(const float* __restrict__ U,
                                       const float* __restrict__ V,
                                       float* __restrict__ out) {
  // Contiguous per-input chunk buffers: LDS chunk stride == global chunk
  // stride == 4096 B, required by the shared-immediate addressing above.
  __shared__ float4 sU[NCHUNK][TPB];    // 16 KB
  __shared__ float4 sV[NCHUNK][TPB];    // 16 KB
  __shared__ float  red[TPB / 32][NACC];
  __shared__ float  fin[NACC];

  const int t = threadIdx.x;
  const int row = blockIdx.x;
  const size_t rowbase = (size_t)row * DDIM;
  const void* baseU = (const void*)(U + rowbase);   // uniform -> SGPR pair
  const void* baseV = (const void*)(V + rowbase);

  const uint32_t lU   = (uint32_t)(uintptr_t)&sU[0][t];  // low 32 bits of a
  const uint32_t lV   = (uint32_t)(uintptr_t)&sV[0][t];  // generic LDS ptr ==
  const uint32_t voff = (uint32_t)t * 16u;               // LDS byte offset

  // Issue the whole row's prefetch up front: 8 async instructions in flight,
  // no address VALU in the drain loop. Completion is FIFO per wave.
  async_copy_b128<0 * CHUNK_B>(lU, voff, baseU);
  async_copy_b128<0 * CHUNK_B>(lV, voff, baseV);
  async_copy_b128<1 * CHUNK_B>(lU, voff, baseU);
  async_copy_b128<1 * CHUNK_B>(lV, voff, baseV);
  async_copy_b128<2 * CHUNK_B>(lU, voff, baseU);
  async_copy_b128<2 * CHUNK_B>(lV, voff, baseV);
  async_copy_b128<3 * CHUNK_B>(lU, voff, baseU);
  async_copy_b128<3 * CHUNK_B>(lV, voff, baseV);

  float acc[NACC];
  #pragma unroll
  for (int i = 0; i < NACC; ++i) acc[i] = 0.0f;

  // Drain: each lane reads back exactly its own LDS slots -> no barriers.
  wait_async<6>(); accum4(acc, sU[0][t], sV[0][t]);
  wait_async<4>(); accum4(acc, sU[1][t], sV[1][t]);
  wait_async<2>(); accum4(acc, sU[2][t], sV[2][t]);
  wait_async<0>(); accum4(acc, sU[3][t], sV[3][t]);

  // Wave32 butterfly reduction (sum; max for chebyshev at index 3).
  #pragma unroll
  for (int m = 16; m > 0; m >>= 1) {
    #pragma unroll
    for (int i = 0; i < NACC; ++i) {
      const float o = __shfl_xor(acc[i], m, 32);
      acc[i] = (i == 3) ? fmaxf(acc[i], o) : (acc[i] + o);
    }
  }

  const int lane = t & 31;
  const int wv   = t >> 5;
  if (lane == 0) {
    #pragma unroll
    for (int i = 0; i < NACC; ++i) red[wv][i] = acc[i];
  }
  __syncthreads();

  if (t < NACC) {
    float a = red[0][t];
    #pragma unroll
    for (int w = 1; w < TPB / 32; ++w) {
      const float b2 = red[w][t];
      a = (t == 3) ? fmaxf(a, b2) : (a + b2);
    }
    fin[t] = a;
  }
  __syncthreads();

  if (t == 0) {
    const float Df   = (float)DDIM;
    const float invD = 1.0f / Df;
    const float s_ad = fin[0], s_abs = fin[1], can = fin[2], cheb = fin[3];
    const float su = fin[4], sv = fin[5], suu = fin[6], svv = fin[7], suv = fin[8];
    const float sdd = fin[9], ham = fin[10];

    const float bray = s_ad / s_abs;

    // correlation: centered moments from raw moments
    const float cnum = suv - su * sv * invD;
    const float um2  = suu - su * su * invD;
    const float vm2  = svv - sv * sv * invD;
    const float corr = 1.0f - cnum / sqrtf(um2 * vm2);

    const float cosd = 1.0f - suv / sqrtf(suu * svv);

    // dice / yule with scipy float semantics
    const float ntt = suv;
    const float ntf = su - suv;
    const float nft = sv - suv;
    const float nff = Df - su - sv + suv;
    const float dice = (ntf + nft) / (2.0f * ntt + ntf + nft);
    const float halfR = ntf * nft;
    const float yden  = ntt * nff + halfR;
    const float yule  = (halfR == 0.0f) ? 0.0f
                        : (2.0f * halfR / ((yden == 0.0f) ? 1.0f : yden));

    const float eucl = sqrtf(sdd);

    float* o = out + (size_t)row * 12;
    o[0]  = bray;        // braycurtis
    o[1]  = can;         // canberra
    o[2]  = cheb;        // chebyshev
    o[3]  = s_ad;        // cityblock
    o[4]  = corr;        // correlation
    o[5]  = cosd;        // cosine
    o[6]  = dice;        // dice
    o[7]  = eucl;        // euclidean
    o[8]  = ham * invD;  // hamming
    o[9]  = eucl;        // minkowski (p=2)
    o[10] = sdd;         // sqeuclidean
    o[11] = yule;        // yule
  }
}

extern "C" void kernel_launch(void* const* d_in, const int* in_sizes, int n_in,
                              void* d_out, int out_size, void* d_ws, size_t ws_size,
                              hipStream_t stream) {
  (void)n_in; (void)out_size; (void)d_ws; (void)ws_size;
  const float* U = (const float*)d_in[0];
  const float* V = (const float*)d_in[1];
  float* out = (float*)d_out;
  const int N = in_sizes[0] / DDIM;   // 8192 rows
  AllDistance_46282567581897_kernel<<<dim3(N), dim3(TPB), 0, stream>>>(U, V, out);
}